// SpatialGCNLayer_72361609003253
// MI455X (gfx1250) — compile-verified
//
#include <hip/hip_runtime.h>

typedef __attribute__((ext_vector_type(2))) float v2f;
typedef __attribute__((ext_vector_type(8))) float v8f;

#define NODES_M   100000
#define IN_F      256
#define OUT_F     256
#define EDGES_E   3200000

#define GEMM_WAVES 8          // waves per block -> 128 rows per block
#define TILE_N     64         // N columns per block
#define PITCH      260        // padded row pitch (floats) for transposed W tile in LDS

// ---------------------------------------------------------------------------
// Kernel 1: support = x @ W  via V_WMMA_F32_16X16X4_F32
// Each wave computes a 16(M) x 64(N) strip; block covers 128(M) x 64(N).
// W[:, n0:n0+64] is staged transposed in LDS so B fragments are ds_load_b64.
// ---------------------------------------------------------------------------
__global__ __launch_bounds__(256) void gcn_gemm_wmma(
    const float* __restrict__ x, const float* __restrict__ w,
    float* __restrict__ support, int n_nodes) {
  extern __shared__ float wt[];   // [TILE_N][PITCH] transposed W panel

  const int tid = threadIdx.x;
  const int n0  = blockIdx.y * TILE_N;

  // Cooperative staging: wt[n][k] = W[k][n0+n]  (coalesced global reads)
  for (int i = tid; i < IN_F * TILE_N; i += 256) {
    int k = i >> 6;       // 0..255
    int n = i & 63;       // 0..63
    wt[n * PITCH + k] = w[k * OUT_F + n0 + n];
  }
  __syncthreads();

  const int wid  = tid >> 5;
  const int lane = tid & 31;
  const int l16  = lane & 15;
  const int hi   = lane >> 4;           // lane half selects K pair / M block
  const int m0   = (blockIdx.x * GEMM_WAVES + wid) * 16;
  if (m0 >= n_nodes) return;            // uniform per-wave guard (EXEC stays full)

  v8f c0 = {}, c1 = {}, c2 = {}, c3 = {};

  // A 16x4 f32 layout: lane l16 = row; v[0]=K(2*hi), v[1]=K(2*hi+1)
  const float* arow = x  + (size_t)(m0 + l16) * IN_F + 2 * hi;
  // B 4x16 f32 layout: lane l16 = col; v[0]=K(2*hi), v[1]=K(2*hi+1)
  const float* bcol = wt + l16 * PITCH + 2 * hi;

#pragma unroll 8
  for (int k = 0; k < IN_F; k += 4) {
    v2f a  = *(const v2f*)(arow + k);
    v2f b0 = *(const v2f*)(bcol + 0 * 16 * PITCH + k);
    v2f b1 = *(const v2f*)(bcol + 1 * 16 * PITCH + k);
    v2f b2 = *(const v2f*)(bcol + 2 * 16 * PITCH + k);
    v2f b3 = *(const v2f*)(bcol + 3 * 16 * PITCH + k);
    c0 = __builtin_amdgcn_wmma_f32_16x16x4_f32(false, a, false, b0, (short)0, c0, false, false);
    c1 = __builtin_amdgcn_wmma_f32_16x16x4_f32(false, a, false, b1, (short)0, c1, false, false);
    c2 = __builtin_amdgcn_wmma_f32_16x16x4_f32(false, a, false, b2, (short)0, c2, false, false);
    c3 = __builtin_amdgcn_wmma_f32_16x16x4_f32(false, a, false, b3, (short)0, c3, false, false);
  }

  // C/D 16x16 f32 layout: VGPR r, lanes0-15 -> M=r, lanes16-31 -> M=r+8, N=l16
  float* o = support + (size_t)(m0 + 8 * hi) * OUT_F + n0 + l16;
#pragma unroll
  for (int r = 0; r < 8; ++r) {
    o[(size_t)r * OUT_F + 0]  = c0[r];
    o[(size_t)r * OUT_F + 16] = c1[r];
    o[(size_t)r * OUT_F + 32] = c2[r];
    o[(size_t)r * OUT_F + 48] = c3[r];
  }
}

// ---------------------------------------------------------------------------
// Kernel 2: out[n][f] = bias[f]   (so zero-degree rows are correct and SpMM
// can be pure atomic accumulation)
// ---------------------------------------------------------------------------
__global__ __launch_bounds__(256) void gcn_init_out(
    const float* __restrict__ bias, float* __restrict__ out, int n_nodes) {
  size_t i = (size_t)blockIdx.x * blockDim.x + threadIdx.x;   // over float4s
  size_t total = (size_t)n_nodes * (OUT_F / 4);
  if (i >= total) return;
  int fq = (int)(i & (OUT_F / 4 - 1));
  ((float4*)out)[i] = ((const float4*)bias)[fq];
}

// ---------------------------------------------------------------------------
// Kernel 3: SpMM with sorted-row segment accumulation.
// One wave per 128-edge chunk; lane owns 8 features (2 x float4).
// Running accumulator flushed with global_atomic_add_f32 on row change.
// ---------------------------------------------------------------------------
#define CHUNK 128

__device__ __forceinline__ void flush_acc(float* dst, const float4& a0, const float4& a1) {
  unsafeAtomicAdd(dst + 0, a0.x);
  unsafeAtomicAdd(dst + 1, a0.y);
  unsafeAtomicAdd(dst + 2, a0.z);
  unsafeAtomicAdd(dst + 3, a0.w);
  unsafeAtomicAdd(dst + 4, a1.x);
  unsafeAtomicAdd(dst + 5, a1.y);
  unsafeAtomicAdd(dst + 6, a1.z);
  unsafeAtomicAdd(dst + 7, a1.w);
}

__global__ __launch_bounds__(256) void gcn_spmm(
    const int* __restrict__ erow, const int* __restrict__ ecol,
    const float* __restrict__ eval, const float* __restrict__ support,
    float* __restrict__ out, int n_edges) {
  const long worker = (long)((blockIdx.x * blockDim.x + threadIdx.x) >> 5);
  const int  lane   = threadIdx.x & 31;
  long e0 = worker * CHUNK;
  if (e0 >= n_edges) return;
  long e1 = e0 + CHUNK;
  if (e1 > n_edges) e1 = n_edges;

  const int f = lane * 8;                  // this lane's feature base
  float4 a0 = make_float4(0.f, 0.f, 0.f, 0.f);
  float4 a1 = make_float4(0.f, 0.f, 0.f, 0.f);
  int cur = erow[e0];

  for (long e = e0; e < e1; ++e) {
    int r = erow[e];
    if (r != cur) {
      flush_acc(out + (size_t)cur * OUT_F + f, a0, a1);
      a0 = make_float4(0.f, 0.f, 0.f, 0.f);
      a1 = make_float4(0.f, 0.f, 0.f, 0.f);
      cur = r;
    }
    int   c = ecol[e];
    float v = eval[e];
    const float4* s = (const float4*)(support + (size_t)c * OUT_F + f);
    float4 s0 = s[0], s1 = s[1];
    a0.x = fmaf(v, s0.x, a0.x);  a0.y = fmaf(v, s0.y, a0.y);
    a0.z = fmaf(v, s0.z, a0.z);  a0.w = fmaf(v, s0.w, a0.w);
    a1.x = fmaf(v, s1.x, a1.x);  a1.y = fmaf(v, s1.y, a1.y);
    a1.z = fmaf(v, s1.z, a1.z);  a1.w = fmaf(v, s1.w, a1.w);
  }
  flush_acc(out + (size_t)cur * OUT_F + f, a0, a1);
}

// ---------------------------------------------------------------------------
extern "C" void kernel_launch(void* const* d_in, const int* in_sizes, int n_in,
                              void* d_out, int out_size, void* d_ws, size_t ws_size,
                              hipStream_t stream) {
  const float* x    = (const float*)d_in[0];
  const int*   erow = (const int*)d_in[1];
  const int*   ecol = (const int*)d_in[2];
  const float* eval = (const float*)d_in[3];
  const float* w    = (const float*)d_in[4];
  const float* bias = (const float*)d_in[5];
  float* out = (float*)d_out;

  const int n_nodes = in_sizes[0] / IN_F;     // 100000
  const int n_edges = in_sizes[1];            // 3200000

  float* support = (float*)d_ws;              // n_nodes * OUT_F floats (102.4 MB)

  // 1) support = x @ W  (WMMA f32)
  {
    int mtiles  = (n_nodes + 16 * GEMM_WAVES - 1) / (16 * GEMM_WAVES); // 782
    dim3 grid(mtiles, OUT_F / TILE_N);                                  // (782, 4)
    size_t lds = (size_t)TILE_N * PITCH * sizeof(float);                // 66,560 B
    gcn_gemm_wmma<<<grid, 256, lds, stream>>>(x, w, support, n_nodes);
  }

  // 2) out = bias (broadcast)
  {
    size_t total4 = (size_t)n_nodes * (OUT_F / 4);
    int blocks = (int)((total4 + 255) / 256);
    gcn_init_out<<<blocks, 256, 0, stream>>>(bias, out, n_nodes);
  }

  // 3) out += segment_sum(val * support[col])
  {
    long workers = ((long)n_edges + CHUNK - 1) / CHUNK;   // 25000 waves
    long threads = workers * 32;
    int blocks = (int)((threads + 255) / 256);            // 3125
    gcn_spmm<<<blocks, 256, 0, stream>>>(erow, ecol, eval, support, out, n_edges);
  }
}